// GCN_24060406792745
// MI455X (gfx1250) — compile-verified
//
#include <hip/hip_runtime.h>

typedef __attribute__((ext_vector_type(2))) float v2f;
typedef __attribute__((ext_vector_type(8))) float v8f;

// ---------------------------------------------------------------------------
// GCN(1 -> 8 -> 1) on 1M nodes / 10M edges, reformulated as scalar scatters:
//   deg[d] += 1 per edge;  dinv = rsqrt(deg + 1 /*self-loop*/)
//   s1[d] += dinv[s]*dinv[d]*x[s]              (layer-1 scalar message)
//   h2[n] = sum_j relu(W1[j]*(s1[n]+dinv[n]^2*x[n]) + b1[j]) * W2[j]   (WMMA)
//   out[d] += dinv[s]*dinv[d]*h2[s];  out[n] += dinv[n]^2*h2[n] + b2
// Node arrays (4 MB each) live in the 192MB L2; edge streams use NT hints.
// ---------------------------------------------------------------------------

__global__ void gcn_init(float* __restrict__ deg, float* __restrict__ s1,
                         float* __restrict__ out, int n) {
    int i = blockIdx.x * blockDim.x + threadIdx.x;
    if (i < n) { deg[i] = 0.0f; s1[i] = 0.0f; out[i] = 0.0f; }
}

__global__ void gcn_degree(const int* __restrict__ dst, float* __restrict__ deg, int e) {
    int i = blockIdx.x * blockDim.x + threadIdx.x;
    if (i < e) {
        int d = __builtin_nontemporal_load(&dst[i]);
        atomicAdd(&deg[d], 1.0f);
    }
}

__global__ void gcn_dinv(float* __restrict__ deg, int n) {
    int i = blockIdx.x * blockDim.x + threadIdx.x;
    if (i < n) {
        float d = deg[i] + 1.0f;           // +1 self-loop; always > 0
        deg[i] = rsqrtf(d);                // deg[] becomes dinv[] in place
    }
}

__global__ void gcn_scatter1(const int* __restrict__ src, const int* __restrict__ dst,
                             const float* __restrict__ x, const float* __restrict__ dinv,
                             float* __restrict__ s1, int e) {
    int i = blockIdx.x * blockDim.x + threadIdx.x;
    if (i < e) {
        int s = __builtin_nontemporal_load(&src[i]);
        int d = __builtin_nontemporal_load(&dst[i]);
        atomicAdd(&s1[d], dinv[s] * dinv[d] * x[s]);
    }
}

// Per-node MLP via V_WMMA_F32_16X16X4_F32: 16 nodes per wave.
// D[j,m] = W1pad[j]*s_m + b1pad[j]  (j = row = VGPR index, m = node = lane 0..15)
// then h2[m] = sum_{j<8} relu(D[j,m]) * W2[j]  -- pure per-lane FMAs, no shuffle.
__global__ void gcn_node_mlp(const float* __restrict__ s1, const float* __restrict__ dinv,
                             const float* __restrict__ x,
                             const float* __restrict__ W1, const float* __restrict__ b1,
                             const float* __restrict__ W2,
                             float* __restrict__ h2, int n) {
    int lane = threadIdx.x & 31;
    int wave = threadIdx.x >> 5;
    int base = (blockIdx.x * (blockDim.x >> 5) + wave) * 16;
    int node = base + lane;

    // Per-lane node scalar (lanes 0..15), with self-loop term folded in.
    float sv = 0.0f;
    if (lane < 16 && node < n) {
        float di = dinv[node];
        sv = s1[node] + di * di * x[node];
    }

    // A (16x4, M=rows=j): column K=0 = W1 padded to 16. Lanes 0-15 hold K=0 in a.x.
    v2f a; a.x = 0.0f; a.y = 0.0f;
    if (lane < 8) a.x = W1[lane];
    // B (4x16, N=cols=nodes): row K=0 = node scalars. Lanes 0-15 hold K=0 in b.x.
    v2f b; b.x = (lane < 16) ? sv : 0.0f; b.y = 0.0f;
    // C (16x16): C[j,m] = b1pad[j]. VGPR r: lanes 0-15 -> row r, lanes 16-31 -> row r+8 (=0).
    v8f c;
#pragma unroll
    for (int r = 0; r < 8; ++r) c[r] = (lane < 16) ? b1[r] : 0.0f;

    // EXEC is all-1s here (no surrounding divergence): required for WMMA.
    v8f dacc = __builtin_amdgcn_wmma_f32_16x16x4_f32(
        /*neg_a=*/false, a, /*neg_b=*/false, b,
        /*c_mod=*/(short)0, c, /*reuse_a=*/false, /*reuse_b=*/false);

    // relu + contraction with W2 (uniform scalars -> s_loads), per lane m.
    float acc = 0.0f;
#pragma unroll
    for (int r = 0; r < 8; ++r) acc = fmaf(fmaxf(dacc[r], 0.0f), W2[r], acc);

    if (lane < 16 && node < n) h2[node] = acc;
}

__global__ void gcn_scatter2(const int* __restrict__ src, const int* __restrict__ dst,
                             const float* __restrict__ h2, const float* __restrict__ dinv,
                             float* __restrict__ out, int e) {
    int i = blockIdx.x * blockDim.x + threadIdx.x;
    if (i < e) {
        int s = __builtin_nontemporal_load(&src[i]);
        int d = __builtin_nontemporal_load(&dst[i]);
        atomicAdd(&out[d], dinv[s] * dinv[d] * h2[s]);
    }
}

__global__ void gcn_final(float* __restrict__ out, const float* __restrict__ h2,
                          const float* __restrict__ dinv, const float* __restrict__ b2, int n) {
    int i = blockIdx.x * blockDim.x + threadIdx.x;
    if (i < n) {
        float di = dinv[i];
        out[i] = out[i] + di * di * h2[i] + b2[0];
    }
}

extern "C" void kernel_launch(void* const* d_in, const int* in_sizes, int n_in,
                              void* d_out, int out_size, void* d_ws, size_t ws_size,
                              hipStream_t stream) {
    const float* x  = (const float*)d_in[0];
    const int*   ei = (const int*)d_in[1];    // edge_index as int32 per harness dtype map
    const float* W1 = (const float*)d_in[2];
    const float* b1 = (const float*)d_in[3];
    const float* W2 = (const float*)d_in[4];
    const float* b2 = (const float*)d_in[5];

    const int n = in_sizes[0];       // 1,000,000 nodes
    const int e = in_sizes[1] / 2;   // 10,000,000 edges
    const int* src = ei;             // edge_index[0, :]
    const int* dst = ei + e;         // edge_index[1, :]

    float* deg = (float*)d_ws;       // n floats; becomes dinv in place
    float* s1  = deg + n;            // n floats
    float* h2  = s1 + n;             // n floats (12 MB total workspace)
    float* out = (float*)d_out;      // n floats, used as the layer-2 accumulator

    const int tb = 256;
    const int nb_n = (n + tb - 1) / tb;
    const int nb_e = (e + tb - 1) / tb;

    gcn_init   <<<nb_n, tb, 0, stream>>>(deg, s1, out, n);
    gcn_degree <<<nb_e, tb, 0, stream>>>(dst, deg, e);
    gcn_dinv   <<<nb_n, tb, 0, stream>>>(deg, n);
    gcn_scatter1<<<nb_e, tb, 0, stream>>>(src, dst, x, deg, s1, e);

    const int nodes_per_block = (tb / 32) * 16;   // 16 nodes per wave
    gcn_node_mlp<<<(n + nodes_per_block - 1) / nodes_per_block, tb, 0, stream>>>(
        s1, deg, x, W1, b1, W2, h2, n);

    gcn_scatter2<<<nb_e, tb, 0, stream>>>(src, dst, h2, deg, out, e);
    gcn_final  <<<nb_n, tb, 0, stream>>>(out, h2, deg, b2, n);
}